// GBlock_481036337798
// MI455X (gfx1250) — compile-verified
//
#include <hip/hip_runtime.h>
#include <math.h>

#define NN 50000
#define CC 256
#define EE 800000
#define KK 512            // concat(agg, x) along K
#define LDSS 516          // LDS row stride in floats: 516 % 64 == 4 -> conflict-free A reads
#define EPSV 1e-5f

typedef __attribute__((ext_vector_type(2))) float v2f;
typedef __attribute__((ext_vector_type(8))) float v8f;
typedef __attribute__((address_space(3))) float lds_float;

// ---------------- zero workspace (agg + sum + sumsq) ----------------
__global__ void k_zero(float4* __restrict__ p, int n4) {
    int i = blockIdx.x * blockDim.x + threadIdx.x;
    if (i < n4) p[i] = make_float4(0.f, 0.f, 0.f, 0.f);
}

// ---------------- build K-pair-packed B: BtP[k2][c] = {Wcat[c][2k2], Wcat[c][2k2+1]} ----------------
__global__ void k_build_bt(const float* __restrict__ Wrel, const float* __restrict__ Wroot,
                           v2f* __restrict__ BtP) {
    int idx = blockIdx.x * blockDim.x + threadIdx.x;   // (KK/2)*CC threads
    int k2 = idx >> 8;
    int c  = idx & 255;
    int k  = k2 * 2;                                   // k and k+1 in same source matrix
    const float* W = (k < CC) ? (Wrel + (size_t)c * CC + k)
                              : (Wroot + (size_t)c * CC + (k - CC));
    v2f v; v.x = W[0]; v.y = W[1];
    BtP[idx] = v;
}

// ---------------- edge scatter: agg[dst] += ew * x[src], one wave per edge ----------------
__global__ __launch_bounds__(256) void k_scatter(const float* __restrict__ x,
                                                 const float* __restrict__ ew,
                                                 const int* __restrict__ ei,
                                                 float* __restrict__ agg) {
    int wave = threadIdx.x >> 5;
    int lane = threadIdx.x & 31;
    int e = blockIdx.x * 8 + wave;           // EE/8 blocks, exact
    int src = ei[e];
    int dst = ei[EE + e];
    float w = ew[e];
    const float4* xs = (const float4*)(x + (size_t)src * CC);
    float4 a = xs[lane];        // channels 4*lane .. 4*lane+3
    float4 b = xs[32 + lane];   // channels 128+4*lane ..
    float* ad = agg + (size_t)dst * CC + lane * 4;
    atomicAdd(ad + 0,       a.x * w);
    atomicAdd(ad + 1,       a.y * w);
    atomicAdd(ad + 2,       a.z * w);
    atomicAdd(ad + 3,       a.w * w);
    atomicAdd(ad + 128 + 0, b.x * w);
    atomicAdd(ad + 128 + 1, b.y * w);
    atomicAdd(ad + 128 + 2, b.z * w);
    atomicAdd(ad + 128 + 3, b.w * w);
}

// ---------------- WMMA f32 GEMM: out = [agg|x] @ B + b_rel ----------------
// Block: 256 thr = 8 waves. Block computes 16 rows x 256 cols; wave w owns cols [32w, 32w+32).
__global__ __launch_bounds__(256) void k_gemm(const float* __restrict__ agg,
                                              const float* __restrict__ x,
                                              const v2f* __restrict__ BtP,
                                              const float* __restrict__ brel,
                                              float* __restrict__ out) {
    __shared__ float Atile[16 * LDSS];
    const int tid = threadIdx.x;
    const int r0  = blockIdx.x * 16;

    // Async direct-to-LDS staging of the 16x512 fp32 A-tile (agg rows then x rows along K).
    // GLOBAL_LOAD_ASYNC_TO_LDS_B128, GV mode: per-lane 64-bit global addr + per-lane LDS offset.
    #pragma unroll
    for (int i = 0; i < 8; ++i) {
        int idx = tid + i * 256;          // float4 index in tile, 2048 total
        int m   = idx >> 7;               // 128 float4 per row
        int k   = (idx & 127) * 4;
        const float* srcp = (k < CC) ? (agg + (size_t)(r0 + m) * CC + k)
                                     : (x   + (size_t)(r0 + m) * CC + (k - CC));
        unsigned long long gaddr = (unsigned long long)srcp;
        unsigned lds_off = (unsigned)(uintptr_t)(lds_float*)&Atile[m * LDSS + k];
        asm volatile("global_load_async_to_lds_b128 %0, %1, off"
                     :: "v"(lds_off), "v"(gaddr) : "memory");
    }
    asm volatile("s_wait_asynccnt 0x0" ::: "memory");
    __syncthreads();

    const int wave = tid >> 5;
    const int lane = tid & 31;
    const int half = lane >> 4;           // 0: lanes 0-15, 1: lanes 16-31
    const int l16  = lane & 15;
    const int cbase = wave * 32;
    const int koff  = half * 2;           // A/B VGPR0 holds K=0 (lo lanes) / K=2 (hi lanes)

    v8f acc0 = {};
    v8f acc1 = {};
    const float* arow = &Atile[l16 * LDSS + koff];
    const v2f* bp0 = BtP + cbase + l16;        // + k2*CC walks K-pairs
    const v2f* bp1 = BtP + cbase + 16 + l16;

    #pragma unroll 8
    for (int k0 = 0; k0 < KK; k0 += 4) {
        v2f a;
        a.x = arow[k0];                   // A[M=l16][K=k0+koff]
        a.y = arow[k0 + 1];               // A[M=l16][K=k0+koff+1]
        const int k2 = (k0 + koff) >> 1;  // kk is even; one b64 load per B fragment
        v2f b0 = bp0[(size_t)k2 * CC];
        v2f b1 = bp1[(size_t)k2 * CC];
        acc0 = __builtin_amdgcn_wmma_f32_16x16x4_f32(false, a, false, b0,
                                                     (short)0, acc0, false, false);
        acc1 = __builtin_amdgcn_wmma_f32_16x16x4_f32(false, a, false, b1,
                                                     (short)0, acc1, false, false);
    }

    // D layout: VGPR r holds M=r (lanes 0-15) / M=r+8 (lanes 16-31), N = lane%16
    const int row_off = half * 8;
    const int col0 = cbase + l16;
    const int col1 = cbase + 16 + l16;
    const float bias0 = brel[col0];
    const float bias1 = brel[col1];
    #pragma unroll
    for (int r = 0; r < 8; ++r) {
        const size_t row = (size_t)(r0 + row_off + r);
        out[row * CC + col0] = acc0[r] + bias0;
        out[row * CC + col1] = acc1[r] + bias1;
    }
}

// ---------------- per-feature stats over nodes ----------------
__global__ __launch_bounds__(256) void k_stats(const float* __restrict__ out,
                                               float* __restrict__ sum,
                                               float* __restrict__ sumsq) {
    int c = threadIdx.x;
    int rbeg = blockIdx.x * 256;
    int rend = rbeg + 256; if (rend > NN) rend = NN;
    float s = 0.f, s2 = 0.f;
    for (int r = rbeg; r < rend; ++r) {
        float v = out[(size_t)r * CC + c];
        s += v; s2 += v * v;
    }
    atomicAdd(&sum[c], s);
    atomicAdd(&sumsq[c], s2);
}

// ---------------- fold BN stats into scale/shift ----------------
__global__ void k_finalize(const float* __restrict__ sum, const float* __restrict__ sumsq,
                           const float* __restrict__ gamma, const float* __restrict__ beta,
                           float* __restrict__ scale, float* __restrict__ shift) {
    int c = threadIdx.x;
    float mean = sum[c] * (1.0f / (float)NN);
    float var  = sumsq[c] * (1.0f / (float)NN) - mean * mean;
    float inv  = rsqrtf(var + EPSV);
    float sc   = gamma[c] * inv;
    scale[c] = sc;
    shift[c] = beta[c] - mean * sc;
}

// ---------------- normalize + relu + skip + relu, in place on d_out ----------------
__global__ __launch_bounds__(256) void k_epilogue(float* __restrict__ out,
                                                  const float* __restrict__ x,
                                                  const float* __restrict__ scale,
                                                  const float* __restrict__ shift) {
    int i = blockIdx.x * 256 + threadIdx.x;   // float4 index, exact grid
    int c4 = i & 63;                          // 64 float4 per row of 256 channels
    float4 sc = ((const float4*)scale)[c4];
    float4 sh = ((const float4*)shift)[c4];
    float4 o  = ((float4*)out)[i];
    float4 xv = ((const float4*)x)[i];
    o.x = fmaxf(fmaxf(o.x * sc.x + sh.x, 0.f) + xv.x, 0.f);
    o.y = fmaxf(fmaxf(o.y * sc.y + sh.y, 0.f) + xv.y, 0.f);
    o.z = fmaxf(fmaxf(o.z * sc.z + sh.z, 0.f) + xv.z, 0.f);
    o.w = fmaxf(fmaxf(o.w * sc.w + sh.w, 0.f) + xv.w, 0.f);
    ((float4*)out)[i] = o;
}

extern "C" void kernel_launch(void* const* d_in, const int* in_sizes, int n_in,
                              void* d_out, int out_size, void* d_ws, size_t ws_size,
                              hipStream_t stream) {
    const float* x     = (const float*)d_in[0];
    const float* ew    = (const float*)d_in[1];
    const float* Wrel  = (const float*)d_in[2];
    const float* brel  = (const float*)d_in[3];
    const float* Wroot = (const float*)d_in[4];
    const float* gamma = (const float*)d_in[5];
    const float* beta  = (const float*)d_in[6];
    const int*   ei    = (const int*)d_in[7];
    // d_in[8] = batch (all zeros) unused

    float* out = (float*)d_out;          // pre-norm GEMM output lives here, then final result
    float* ws  = (float*)d_ws;
    float* agg   = ws;                                   // NN*CC
    float* sum   = ws + (size_t)NN * CC;                 // CC
    float* sumsq = sum + CC;                             // CC
    float* scale = sumsq + CC;                           // CC
    float* shift = scale + CC;                           // CC
    v2f*   BtP   = (v2f*)(shift + CC);                   // (KK/2)*CC float2 pairs

    const int zero4 = ((size_t)NN * CC + 2 * CC) / 4;    // agg + sum + sumsq (contiguous)
    k_zero    <<<(zero4 + 255) / 256,   256, 0, stream>>>((float4*)ws, zero4);
    k_build_bt<<<(KK / 2 * CC) / 256,   256, 0, stream>>>(Wrel, Wroot, BtP);
    k_scatter <<<EE / 8,                256, 0, stream>>>(x, ew, ei, agg);
    k_gemm    <<<NN / 16,               256, 0, stream>>>(agg, x, BtP, brel, out);
    k_stats   <<<(NN + 255) / 256,      256, 0, stream>>>(out, sum, sumsq);
    k_finalize<<<1,                     256, 0, stream>>>(sum, sumsq, gamma, beta, scale, shift);
    k_epilogue<<<(NN * CC / 4) / 256,   256, 0, stream>>>(out, x, scale, shift);
}